// EMALoss_54674933678414
// MI455X (gfx1250) — compile-verified
//
#include <hip/hip_runtime.h>

#define NUM_CLASS 1000
#define DIMS      2048
#define HEAD      256
#define ALPHA     0.999f

typedef __attribute__((ext_vector_type(2))) float v2f;
typedef __attribute__((ext_vector_type(8))) float v8f;

// ---------------------------------------------------------------- init last[]
__global__ void ema_init_last(int* last) {
    int i = blockIdx.x * blockDim.x + threadIdx.x;
    if (i < NUM_CLASS) last[i] = -1;
}

// ------------------------------------------------- argmax over duplicate labels
// last[c] = max{ n : labels[n] == c }  (deterministic; int atomicMax)
__global__ void ema_argmax(const int* __restrict__ labels, int* __restrict__ last, int n) {
    int i = blockIdx.x * blockDim.x + threadIdx.x;
    if (i < n) atomicMax(&last[labels[i]], i);
}

// ---------------------------------------------------------------- loss partials
// Each wave owns a 16-row tile of x. Per iteration a lane produces two squared
// diffs placed per the 16x4 f32 A-matrix layout (lanes 0-15: K=0,1; lanes
// 16-31: K=2,3) and accumulates row sums via V_WMMA_F32_16X16X4_F32 with an
// all-ones B. After the loop every D column holds the 16 row sums, so the
// full lane/VGPR reduction equals 16x the tile total (scaled out at the end).
__global__ void ema_loss_partial(const float* __restrict__ x,
                                 const int*   __restrict__ labels,
                                 const float* __restrict__ centers,
                                 float*       __restrict__ partials) {
    __shared__ float wsum[8];
    const int lane = threadIdx.x & 31;
    const int wave = threadIdx.x >> 5;
    const int tile = blockIdx.x * 8 + wave;      // 16-row tile id
    const int row  = tile * 16 + (lane & 15);
    const int half = lane >> 4;                  // 0 -> cols {0,1}, 1 -> cols {2,3}

    const int lbl = labels[row];
    const float2* __restrict__ px = (const float2*)(x       + (size_t)row * DIMS);
    const float2* __restrict__ pc = (const float2*)(centers + (size_t)lbl * DIMS);

    v8f acc = {};
    v2f bones; bones[0] = 1.0f; bones[1] = 1.0f;

#pragma unroll 4
    for (int k = 0; k < DIMS / 4; ++k) {
        float2 a = px[2 * k + half];
        float2 c = pc[2 * k + half];
        float dx = a.x - c.x;
        float dy = a.y - c.y;
        v2f sq; sq[0] = dx * dx; sq[1] = dy * dy;
        // D = A(16x4 sq-diffs) * B(4x16 ones) + C  -> per-row running sums
        acc = __builtin_amdgcn_wmma_f32_16x16x4_f32(false, sq, false, bones,
                                                    (short)0, acc, false, false);
    }

    float s = acc[0] + acc[1] + acc[2] + acc[3] + acc[4] + acc[5] + acc[6] + acc[7];
    s += __shfl_xor(s, 16, 32);
    s += __shfl_xor(s,  8, 32);
    s += __shfl_xor(s,  4, 32);
    s += __shfl_xor(s,  2, 32);
    s += __shfl_xor(s,  1, 32);
    if (lane == 0) wsum[wave] = s;
    __syncthreads();
    if (threadIdx.x == 0) {
        float b = 0.0f;
#pragma unroll
        for (int i = 0; i < 8; ++i) b += wsum[i];
        partials[blockIdx.x] = b;   // == 16 * sum of squared diffs of this block
    }
}

// ------------------------------------------------------------- final reduction
__global__ void ema_loss_reduce(const float* __restrict__ partials, int nblocks,
                                float* __restrict__ out, float scale) {
    __shared__ float sm[128];
    int t = threadIdx.x;
    float s = 0.0f;
    for (int i = t; i < nblocks; i += 128) s += partials[i];
    sm[t] = s;
    __syncthreads();
    for (int off = 64; off > 0; off >>= 1) {
        if (t < off) sm[t] += sm[t + off];
        __syncthreads();
    }
    if (t == 0) out[0] = sm[0] * scale;
}

// ------------------------------------------------------------- centers update
// Vectorized f4 loads; scalar stores because d_out+1 is only 4-byte aligned.
__global__ void ema_centers_update(const float* __restrict__ x,
                                   const float* __restrict__ centers,
                                   const int*   __restrict__ last,
                                   float*       __restrict__ outc) {
    int i   = blockIdx.x * blockDim.x + threadIdx.x;   // float4 index
    int cls = i >> 9;                                  // 512 float4 per class row
    int j4  = i & 511;
    if (cls >= NUM_CLASS) return;

    float4 cv = ((const float4*)centers)[i];
    int ln = last[cls];
    float4 o;
    if (ln >= 0) {
        float4 xv = ((const float4*)(x + (size_t)ln * DIMS))[j4];
        o.x = ALPHA * cv.x + (1.0f - ALPHA) * xv.x;
        o.y = ALPHA * cv.y + (1.0f - ALPHA) * xv.y;
        o.z = ALPHA * cv.z + (1.0f - ALPHA) * xv.z;
        o.w = ALPHA * cv.w + (1.0f - ALPHA) * xv.w;
    } else {
        o = cv;
    }
    size_t base = (size_t)i * 4;
    outc[base + 0] = o.x;
    outc[base + 1] = o.y;
    outc[base + 2] = o.z;
    outc[base + 3] = o.w;
}

extern "C" void kernel_launch(void* const* d_in, const int* in_sizes, int n_in,
                              void* d_out, int out_size, void* d_ws, size_t ws_size,
                              hipStream_t stream) {
    const float* x       = (const float*)d_in[0];
    const int*   labels  = (const int*)  d_in[1];
    const float* centers = (const float*)d_in[2];
    float*       out     = (float*)d_out;

    int*   last     = (int*)d_ws;
    float* partials = (float*)((char*)d_ws + 4096);

    const int n = in_sizes[1];                 // 16384 samples
    const int loss_blocks = n / 128;           // 8 waves x 16 rows per block

    ema_init_last<<<(NUM_CLASS + 255) / 256, 256, 0, stream>>>(last);
    ema_argmax<<<(n + 255) / 256, 256, 0, stream>>>(labels, last, n);
    ema_loss_partial<<<loss_blocks, 256, 0, stream>>>(x, labels, centers, partials);
    // partials carry 16x the true sum (16 identical D columns) -> fold into scale
    const float scale = 1.0f / (16.0f * (float)n * (float)HEAD);
    ema_loss_reduce<<<1, 128, 0, stream>>>(partials, loss_blocks, out, scale);

    const int upd_elems  = NUM_CLASS * DIMS / 4;   // float4 count
    ema_centers_update<<<(upd_elems + 255) / 256, 256, 0, stream>>>(x, centers, last, out + 1);
}